// DQGNN_layer_31112743092862
// MI455X (gfx1250) — compile-verified
//
#include <hip/hip_runtime.h>
#include <math.h>
#include <stdint.h>

typedef __attribute__((ext_vector_type(2))) float v2f;
typedef __attribute__((ext_vector_type(8))) float v8f;

#define HALF_F 128

// ---------------------------------------------------------------------------
// Build W = [[A_h, B_h], [0, A_h]] (quaternion-expanded), stored in a
// WMMA-B-friendly layout: 8 strips of 32 columns; within a strip, element
// (k, nc) lives at ((k>>1)*32 + nc)*2 + (k&1)  -> B fragments become single
// aligned ds_load_b64 into a VGPR pair, and strip staging is contiguous.
// ---------------------------------------------------------------------------
__global__ void build_w_kernel(const float* __restrict__ A,
                               const float* __restrict__ B,
                               float* __restrict__ Wb) {
  const int o = threadIdx.x;   // 0..255 (output feature)
  const int i = blockIdx.x;    // 0..255 (input feature / K)
  const int   comp[4][4] = {{0,1,2,3},{1,0,3,2},{2,3,0,1},{3,2,1,0}};
  const float sgn [4][4] = {{1.f,-1.f,-1.f,-1.f},
                            {1.f, 1.f,-1.f, 1.f},
                            {1.f, 1.f, 1.f,-1.f},
                            {1.f,-1.f, 1.f, 1.f}};
  float v;
  if (i >= HALF_F && o < HALF_F) {
    v = 0.0f;                                   // bottom-left block is zero
  } else {
    const float* K = (i < HALF_F && o >= HALF_F) ? B : A;
    const int r = i & 127, c = o & 127;
    const int rb = r >> 5, cb = c >> 5;
    v = sgn[cb][rb] * K[(r & 31) * 128 + comp[cb][rb] * 32 + (c & 31)];
  }
  const int strip = o >> 5, nc = o & 31;
  Wb[(size_t)strip * 8192 + ((((i >> 1) * 32 + nc) << 1) | (i & 1))] = v;
}

// ---------------------------------------------------------------------------
// support = X @ W via V_WMMA_F32_16X16X4_F32.
// Block: 256 threads = 8 waves; tile 128 rows x 32 cols.
// 32KB W strip async-staged to LDS (contiguous b128 stream), shared by waves.
// ---------------------------------------------------------------------------
__global__ void __launch_bounds__(256)
qgemm_kernel(const float* __restrict__ X,    // n_rows x 256
             const float* __restrict__ Wb,   // interleaved strips
             float* __restrict__ S,          // padded_rows x 256
             int n_rows) {
  __shared__ float Wlds[8192];               // 32 KB
  const int tid = threadIdx.x;
  const float* strip = Wb + (size_t)blockIdx.y * 8192;

  {
    const unsigned ldsbase = (unsigned)(uintptr_t)(&Wlds[0]);
    for (int idx4 = tid; idx4 < 2048; idx4 += 256) {
      const float* gp = strip + idx4 * 4;
      const unsigned la = ldsbase + (unsigned)idx4 * 16u;
      asm volatile("global_load_async_to_lds_b128 %0, %1, off"
                   :: "v"(la), "v"(gp) : "memory");
    }
    asm volatile("s_wait_asynccnt 0" ::: "memory");
  }
  __syncthreads();

  const int wave = tid >> 5;
  const int lane = tid & 31;
  const int lrow = lane & 15;
  const int lhi  = lane >> 4;
  const int mbase = blockIdx.x * 128 + wave * 16;

  int mload = mbase + lrow;
  if (mload >= n_rows) mload = n_rows - 1;     // clamp; pad rows never consumed
  const float* xrow = X + (size_t)mload * 256;

  v8f acc0 = {};
  v8f acc1 = {};

#pragma unroll 4
  for (int k0 = 0; k0 < 256; k0 += 4) {
    // A 16x4 frag: lanes 0-15 -> K=k0..k0+1, lanes 16-31 -> K=k0+2..k0+3
    const v2f a = *(const v2f*)(xrow + k0 + (lhi << 1));
    // B 4x16 frags: K-pair (k0+2*lhi, +1) interleaved -> single b64 each
    const int kp = (k0 >> 1) + lhi;            // K-pair index
    const v2f b0 = *(const v2f*)(Wlds + ((kp * 32 + lrow) << 1));
    const v2f b1 = *(const v2f*)(Wlds + ((kp * 32 + 16 + lrow) << 1));
    acc0 = __builtin_amdgcn_wmma_f32_16x16x4_f32(false, a, false, b0,
                                                 (short)0, acc0, false, false);
    acc1 = __builtin_amdgcn_wmma_f32_16x16x4_f32(false, a, false, b1,
                                                 (short)0, acc1, false, false);
  }

  const int n0 = blockIdx.y * 32;
#pragma unroll
  for (int r = 0; r < 8; ++r) {
    const int m = mbase + r + lhi * 8;
    float* d = S + (size_t)m * 256 + n0;
    d[lrow]      = acc0[r];
    d[16 + lrow] = acc1[r];
  }
}

// ---------------------------------------------------------------------------
// CSR-by-destination build: count -> hierarchical exclusive scan -> fill.
// ---------------------------------------------------------------------------
__global__ void __launch_bounds__(256)
count_kernel(const int* __restrict__ erow, int* __restrict__ deg, int n_edges) {
  const int e = blockIdx.x * 256 + threadIdx.x;
  if (e < n_edges) atomicAdd(&deg[erow[e]], 1);
}

__global__ void __launch_bounds__(1024)
deg_reduce_kernel(const int* __restrict__ deg, int* __restrict__ bsum, int n) {
  __shared__ int buf[1024];
  const int t = threadIdx.x;
  const int i = blockIdx.x * 1024 + t;
  buf[t] = (i < n) ? deg[i] : 0;
  __syncthreads();
  for (int off = 512; off > 0; off >>= 1) {
    if (t < off) buf[t] += buf[t + off];
    __syncthreads();
  }
  if (t == 0) bsum[blockIdx.x] = buf[0];
}

__global__ void __launch_bounds__(1024)
scan_bsum_kernel(int* __restrict__ bsum, int nb) {
  __shared__ int buf[1024];
  const int t = threadIdx.x;
  const int v = (t < nb) ? bsum[t] : 0;
  buf[t] = v;
  __syncthreads();
  for (int off = 1; off < 1024; off <<= 1) {
    const int x = (t >= off) ? buf[t - off] : 0;
    __syncthreads();
    buf[t] += x;
    __syncthreads();
  }
  if (t < nb) bsum[t] = buf[t] - v;           // exclusive, in place
}

__global__ void __launch_bounds__(1024)
scan_finalize_kernel(const int* __restrict__ deg, const int* __restrict__ bsum,
                     int* __restrict__ row_start, int* __restrict__ cursor,
                     int n, int n_edges) {
  __shared__ int buf[1024];
  const int t = threadIdx.x;
  const int i = blockIdx.x * 1024 + t;
  const int v = (i < n) ? deg[i] : 0;
  buf[t] = v;
  __syncthreads();
  for (int off = 1; off < 1024; off <<= 1) {
    const int x = (t >= off) ? buf[t - off] : 0;
    __syncthreads();
    buf[t] += x;
    __syncthreads();
  }
  const int excl = buf[t] - v + bsum[blockIdx.x];
  if (i < n) { row_start[i] = excl; cursor[i] = excl; }
  if (blockIdx.x == 0 && t == 0) row_start[n] = n_edges;
}

__global__ void __launch_bounds__(256)
fill_kernel(const int* __restrict__ erow, int* __restrict__ cursor,
            int* __restrict__ eid, int n_edges) {
  const int e = blockIdx.x * 256 + threadIdx.x;
  if (e < n_edges) {
    const int p = atomicAdd(&cursor[erow[e]], 1);
    eid[p] = e;
  }
}

// ---------------------------------------------------------------------------
// Row gather: one wave per output row, accumulate incident edges in registers,
// write the 1KB row exactly once.  support & the CSR index are L2-resident.
// ---------------------------------------------------------------------------
__global__ void __launch_bounds__(256)
row_gather_kernel(const float* __restrict__ S,
                  const int* __restrict__ row_start,
                  const int* __restrict__ eid,
                  const int* __restrict__ ecol,
                  const float* __restrict__ ew,
                  float* __restrict__ out, int n_rows) {
  const int r = blockIdx.x * 8 + (threadIdx.x >> 5);
  if (r >= n_rows) return;
  const int lane = threadIdx.x & 31;
  const int s0 = row_start[r];
  const int s1 = row_start[r + 1];

  float4 a0 = {0.f, 0.f, 0.f, 0.f};
  float4 a1 = {0.f, 0.f, 0.f, 0.f};
  for (int idx = s0; idx < s1; ++idx) {
    const int e   = eid[idx];
    const int col = ecol[e];
    const float w = ew[e];
    const float4* src = (const float4*)(S + (size_t)col * 256);
    const float4 v0 = src[lane];
    const float4 v1 = src[lane + 32];
    a0.x += v0.x * w; a0.y += v0.y * w; a0.z += v0.z * w; a0.w += v0.w * w;
    a1.x += v1.x * w; a1.y += v1.y * w; a1.z += v1.z * w; a1.w += v1.w * w;
  }
  float4* dst = (float4*)(out + (size_t)r * 256);
  dst[lane]      = a0;
  dst[lane + 32] = a1;
}

// ---------------------------------------------------------------------------
// BN stats + in-place normalize/tanh.
// ---------------------------------------------------------------------------
__global__ void __launch_bounds__(256)
bn_stats_kernel(const float* __restrict__ out,
                float* __restrict__ sums, float* __restrict__ sqs, int n_rows) {
  const int f = threadIdx.x;
  float s = 0.f, q = 0.f;
  for (int r = blockIdx.x; r < n_rows; r += gridDim.x) {
    const float v = out[(size_t)r * 256 + f];
    s += v;
    q += v * v;
  }
  atomicAdd(&sums[f], s);
  atomicAdd(&sqs[f], q);
}

__global__ void __launch_bounds__(256)
bn_apply_kernel(float* __restrict__ out,
                const float* __restrict__ sums, const float* __restrict__ sqs,
                const float* __restrict__ gamma, const float* __restrict__ beta,
                int n_rows) {
  const size_t i = (size_t)blockIdx.x * 256 + threadIdx.x;
  if (i >= (size_t)n_rows * 256) return;
  const int f = (int)(i & 255);
  const float inv_n = 1.0f / (float)n_rows;
  const float mean = sums[f] * inv_n;
  const float var  = sqs[f] * inv_n - mean * mean;
  const float y = (out[i] - mean) * rsqrtf(var + 1e-5f) * gamma[f] + beta[f];
  out[i] = tanhf(y);
}

// ---------------------------------------------------------------------------
extern "C" void kernel_launch(void* const* d_in, const int* in_sizes, int n_in,
                              void* d_out, int out_size, void* d_ws, size_t ws_size,
                              hipStream_t stream) {
  const float* X     = (const float*)d_in[0];
  const float* A     = (const float*)d_in[1];
  const float* B     = (const float*)d_in[2];
  const int*   erow  = (const int*)d_in[3];
  const int*   ecol  = (const int*)d_in[4];
  const float* ew    = (const float*)d_in[5];
  const float* gamma = (const float*)d_in[6];
  const float* beta  = (const float*)d_in[7];

  const int n_rows  = in_sizes[0] / 256;
  const int n_edges = in_sizes[3];
  const int blocks_m = (n_rows + 127) / 128;
  const int nscan   = (n_rows + 1023) / 1024;   // <= 1024

  // ws layout: Wb | S(padded) | sums | sqs | deg | row_start | cursor | bsum | eid
  char* ws = (char*)d_ws;
  float* Wb = (float*)ws;                                   // 256 KB
  float* S  = (float*)(ws + (size_t)65536 * 4);
  const size_t s_bytes = (size_t)blocks_m * 128 * 256 * 4;
  char* p = ws + (size_t)65536 * 4 + s_bytes;
  float* sums      = (float*)p;            p += 256 * 4;
  float* sqs       = (float*)p;            p += 256 * 4;
  int*   deg       = (int*)p;              p += (size_t)n_rows * 4;
  int*   row_start = (int*)p;              p += (size_t)(n_rows + 1) * 4;
  int*   cursor    = (int*)p;              p += (size_t)n_rows * 4;
  int*   bsum      = (int*)p;              p += 1024 * 4;
  int*   eid       = (int*)p;

  // Phase 1: weights + GEMM (support = X @ W)
  build_w_kernel<<<256, 256, 0, stream>>>(A, B, Wb);
  dim3 ggrid(blocks_m, 8);
  qgemm_kernel<<<ggrid, 256, 0, stream>>>(X, Wb, S, n_rows);

  // Phase 2: CSR-by-row build
  hipMemsetAsync(deg, 0, (size_t)n_rows * 4, stream);
  hipMemsetAsync(sums, 0, 2 * 256 * 4, stream);
  count_kernel<<<(n_edges + 255) / 256, 256, 0, stream>>>(erow, deg, n_edges);
  deg_reduce_kernel<<<nscan, 1024, 0, stream>>>(deg, bsum, n_rows);
  scan_bsum_kernel<<<1, 1024, 0, stream>>>(bsum, nscan);
  scan_finalize_kernel<<<nscan, 1024, 0, stream>>>(deg, bsum, row_start, cursor,
                                                   n_rows, n_edges);
  fill_kernel<<<(n_edges + 255) / 256, 256, 0, stream>>>(erow, cursor, eid, n_edges);

  // Phase 3: gather-accumulate per row (no fp atomics)
  row_gather_kernel<<<(n_rows + 7) / 8, 256, 0, stream>>>(S, row_start, eid,
                                                          ecol, ew,
                                                          (float*)d_out, n_rows);

  // Phase 4: batchnorm + tanh
  bn_stats_kernel<<<1024, 256, 0, stream>>>((const float*)d_out, sums, sqs, n_rows);
  const int apply_blocks = (int)(((size_t)n_rows * 256 + 255) / 256);
  bn_apply_kernel<<<apply_blocks, 256, 0, stream>>>((float*)d_out, sums, sqs,
                                                    gamma, beta, n_rows);
}